// NodeGATLayer_57140244906496
// MI455X (gfx1250) — compile-verified
//
#include <hip/hip_runtime.h>
#include <hip/hip_bf16.h>

#define DIM 128

typedef float v2f __attribute__((ext_vector_type(2)));
typedef float v8f __attribute__((ext_vector_type(8)));

__device__ __forceinline__ int f2ord(float f) {
    int i = __float_as_int(f);
    return i >= 0 ? i : (i ^ 0x7FFFFFFF);
}
__device__ __forceinline__ float ord2f(int i) {
    return __int_as_float(i >= 0 ? i : (i ^ 0x7FFFFFFF));
}
__device__ __forceinline__ float leaky(float x) { return x >= 0.f ? x : 0.01f * x; }

// ---------------------------------------------------------------------------
// C[M x 128] = A[M x 128] @ W[128 x 128]^T   (W row-major, out x in)
// A selected among A0/A1/A2 by *sel (device scalar) if sel != nullptr.
// 64 rows per block, 8 waves; each wave owns 16 output columns and loops
// 4 M-tiles; K=128 consumed via 32 chained V_WMMA_F32_16X16X4_F32.
// ---------------------------------------------------------------------------
__global__ __launch_bounds__(256)
void gemm_nt_wmma(const float* __restrict__ A0, const float* __restrict__ A1,
                  const float* __restrict__ A2, const int* __restrict__ sel,
                  const float* __restrict__ W, float* __restrict__ C, int M)
{
    __shared__ float As[64][132];   // stride 132: 16B-aligned rows, banks spread by 4

    const float* A = A0;
    if (sel) {
        int t = *sel;
        A = (t == 1) ? A1 : ((t == 2) ? A2 : A0);
    }

    int brow = blockIdx.x * 64;

    // cooperative A-tile load (float4), zero-pad past M
    for (int t = threadIdx.x; t < 64 * 32; t += 256) {
        int r  = t >> 5;
        int c4 = t & 31;
        int gr = brow + r;
        float4 v = make_float4(0.f, 0.f, 0.f, 0.f);
        if (gr < M) v = ((const float4*)(A + (size_t)gr * DIM))[c4];
        *(float4*)&As[r][c4 * 4] = v;
    }
    __syncthreads();

    int lane = threadIdx.x & 31;
    int wv   = threadIdx.x >> 5;   // 0..7 -> output column tile
    int l16  = lane & 15;
    int half = lane >> 4;          // 0: K pair (k,k+1), 1: K pair (k+2,k+3)
    int jb   = wv * 16;

    // B tile (16 cols of output == 16 rows of W) into registers: 64 floats/lane
    v2f breg[32];
#pragma unroll
    for (int s = 0; s < 32; ++s) {
        int kc = 4 * s + 2 * half;
        breg[s] = *(const v2f*)(W + (size_t)(jb + l16) * DIM + kc);
    }

#pragma unroll
    for (int mt = 0; mt < 4; ++mt) {
        int rb = mt * 16;
        v8f acc = {0.f, 0.f, 0.f, 0.f, 0.f, 0.f, 0.f, 0.f};
#pragma unroll
        for (int s = 0; s < 32; ++s) {
            int kc = 4 * s + 2 * half;
            v2f a = *(const v2f*)&As[rb + l16][kc];
            acc = __builtin_amdgcn_wmma_f32_16x16x4_f32(
                false, a, false, breg[s], (short)0, acc, false, false);
        }
        // C/D layout: VGPR v -> row = rb + v + 8*half, col = jb + l16
#pragma unroll
        for (int v = 0; v < 8; ++v) {
            int row = brow + rb + v + 8 * half;
            if (row < M) C[(size_t)row * DIM + jb + l16] = acc[v];
        }
    }
}

// ---------------------------------------------------------------------------
// sl[n] = h[n] . w2[0:128],  sr[n] = h[n] . w2[128:256]   (wave per node)
// ---------------------------------------------------------------------------
__global__ __launch_bounds__(256)
void row_scores(const float* __restrict__ h, const float* __restrict__ w2,
                float* __restrict__ sl, float* __restrict__ sr, int M)
{
    int n = blockIdx.x * 8 + (threadIdx.x >> 5);
    if (n >= M) return;
    int lane = threadIdx.x & 31;
    float4 hv = ((const float4*)(h + (size_t)n * DIM))[lane];
    float4 wl = ((const float4*)w2)[lane];
    float4 wr = ((const float4*)(w2 + DIM))[lane];
    float pl = hv.x * wl.x + hv.y * wl.y + hv.z * wl.z + hv.w * wl.w;
    float pr = hv.x * wr.x + hv.y * wr.y + hv.z * wr.z + hv.w * wr.w;
#pragma unroll
    for (int off = 16; off; off >>= 1) {
        pl += __shfl_xor(pl, off, 32);
        pr += __shfl_xor(pr, off, 32);
    }
    if (lane == 0) { sl[n] = pl; sr[n] = pr; }
}

// ---------------------------------------------------------------------------
// workspace init: mord = ordered(-inf) sentinel, den = 0, hagg = 0
// ---------------------------------------------------------------------------
__global__ void init_ws_kernel(int* __restrict__ mord, float* __restrict__ den,
                               float* __restrict__ hagg, int n3, long n3d)
{
    long i = (long)blockIdx.x * blockDim.x + threadIdx.x;
    if (i < n3d) hagg[i] = 0.f;
    if (i < n3) { mord[i] = (int)0x80000000u; den[i] = 0.f; }
}

__global__ __launch_bounds__(256)
void edge_max_kernel(const int* __restrict__ src, const int* __restrict__ dst,
                     const float* __restrict__ sl, const float* __restrict__ sr,
                     int* __restrict__ mord, int E)
{
    int i = blockIdx.x * 256 + threadIdx.x;
    if (i >= E) return;
    int d = dst[i];
    float e = leaky(sl[src[i]] + sr[d]);
    atomicMax(&mord[d], f2ord(e));
}

__global__ __launch_bounds__(256)
void edge_expsum_kernel(const int* __restrict__ src, const int* __restrict__ dst,
                        const float* __restrict__ sl, const float* __restrict__ sr,
                        const int* __restrict__ mord, float* __restrict__ den, int E)
{
    int i = blockIdx.x * 256 + threadIdx.x;
    if (i >= E) return;
    int d = dst[i];
    float e = leaky(sl[src[i]] + sr[d]);
    atomicAdd(&den[d], expf(e - ord2f(mord[d])));
}

// wave per edge: hagg[dst] += alpha * N[src]
__global__ __launch_bounds__(256)
void edge_agg_kernel(const int* __restrict__ src, const int* __restrict__ dst,
                     const float* __restrict__ sl, const float* __restrict__ sr,
                     const int* __restrict__ mord, const float* __restrict__ den,
                     const float* __restrict__ Nf, float* __restrict__ hagg, int E)
{
    int i = blockIdx.x * 8 + (threadIdx.x >> 5);
    if (i >= E) return;
    int lane = threadIdx.x & 31;
    int s = src[i], d = dst[i];
    float e = leaky(sl[s] + sr[d]);
    float alpha = expf(e - ord2f(mord[d])) / den[d];
    float4 nv = ((const float4*)(Nf + (size_t)s * DIM))[lane];
    float* o = hagg + (size_t)d * DIM + lane * 4;
    atomicAdd(o + 0, alpha * nv.x);
    atomicAdd(o + 1, alpha * nv.y);
    atomicAdd(o + 2, alpha * nv.z);
    atomicAdd(o + 3, alpha * nv.w);
}

// ---------------------------------------------------------------------------
// per-node: e_r = q.k_r / sqrt(128); softmax over 3; out = sum a_r * h_r
// ---------------------------------------------------------------------------
__global__ __launch_bounds__(256)
void attn_kernel(const float* __restrict__ q, const float* __restrict__ k,
                 const float* __restrict__ hagg, float* __restrict__ out, int M)
{
    int n = blockIdx.x * 8 + (threadIdx.x >> 5);
    if (n >= M) return;
    int lane = threadIdx.x & 31;
    size_t nd = (size_t)M * DIM;
    size_t ro = (size_t)n * DIM;

    float4 qv = ((const float4*)(q + ro))[lane];
    float4 k0 = ((const float4*)(k + ro))[lane];
    float4 k1 = ((const float4*)(k + nd + ro))[lane];
    float4 k2 = ((const float4*)(k + 2 * nd + ro))[lane];

    float e0 = qv.x * k0.x + qv.y * k0.y + qv.z * k0.z + qv.w * k0.w;
    float e1 = qv.x * k1.x + qv.y * k1.y + qv.z * k1.z + qv.w * k1.w;
    float e2 = qv.x * k2.x + qv.y * k2.y + qv.z * k2.z + qv.w * k2.w;
#pragma unroll
    for (int off = 16; off; off >>= 1) {
        e0 += __shfl_xor(e0, off, 32);
        e1 += __shfl_xor(e1, off, 32);
        e2 += __shfl_xor(e2, off, 32);
    }
    const float sc = 0.08838834764831845f;  // 1/sqrt(128)
    e0 *= sc; e1 *= sc; e2 *= sc;
    float mx = fmaxf(e0, fmaxf(e1, e2));
    float x0 = expf(e0 - mx), x1 = expf(e1 - mx), x2 = expf(e2 - mx);
    float inv = 1.f / (x0 + x1 + x2);
    float a0 = x0 * inv, a1 = x1 * inv, a2 = x2 * inv;

    float4 h0 = ((const float4*)(hagg + ro))[lane];
    float4 h1 = ((const float4*)(hagg + nd + ro))[lane];
    float4 h2 = ((const float4*)(hagg + 2 * nd + ro))[lane];
    float4 o;
    o.x = a0 * h0.x + a1 * h1.x + a2 * h2.x;
    o.y = a0 * h0.y + a1 * h1.y + a2 * h2.y;
    o.z = a0 * h0.z + a1 * h1.z + a2 * h2.z;
    o.w = a0 * h0.w + a1 * h1.w + a2 * h2.w;
    ((float4*)(out + ro))[lane] = o;
}

extern "C" void kernel_launch(void* const* d_in, const int* in_sizes, int n_in,
                              void* d_out, int out_size, void* d_ws, size_t ws_size,
                              hipStream_t stream)
{
    const float* Nf    = (const float*)d_in[0];
    const float* W1[3] = {(const float*)d_in[1], (const float*)d_in[2], (const float*)d_in[3]};
    const float* w2[3] = {(const float*)d_in[4], (const float*)d_in[5], (const float*)d_in[6]};
    const float* Wq    = (const float*)d_in[7];
    const float* Wk    = (const float*)d_in[8];
    // d_in[9] = Wv: computed-but-unused in reference output
    const int* srcs[3] = {(const int*)d_in[10], (const int*)d_in[12], (const int*)d_in[14]};
    const int* dsts[3] = {(const int*)d_in[11], (const int*)d_in[13], (const int*)d_in[15]};
    const int* tgt     = (const int*)d_in[16];

    int M = in_sizes[0] / DIM;          // 100000
    long ND = (long)M * DIM;

    // workspace layout (floats)
    float* ws   = (float*)d_ws;
    float* hagg = ws;                   // 3*ND
    float* kbuf = hagg + 3 * ND;        // 3*ND
    float* qbuf = kbuf + 3 * ND;        // ND (doubles as fc1 h scratch)
    float* sl   = qbuf + ND;            // 3*M
    float* sr   = sl + 3L * M;          // 3*M
    float* den  = sr + 3L * M;          // 3*M
    int*   mord = (int*)(den + 3L * M); // 3*M

    long n3d = 3 * ND;
    int  n3  = 3 * M;
    {
        int blocks = (int)((n3d + 255) / 256);
        init_ws_kernel<<<blocks, 256, 0, stream>>>(mord, den, hagg, n3, n3d);
    }

    int gblocks = (M + 63) / 64;

    // per-relation fc1 + node scores
    for (int r = 0; r < 3; ++r) {
        gemm_nt_wmma<<<gblocks, 256, 0, stream>>>(Nf, Nf, Nf, nullptr, W1[r], qbuf, M);
        row_scores<<<(M + 7) / 8, 256, 0, stream>>>(qbuf, w2[r], sl + (long)r * M,
                                                    sr + (long)r * M, M);
    }

    // segment softmax over edges (max, expsum, aggregate)
    for (int r = 0; r < 3; ++r) {
        int E = in_sizes[10 + 2 * r];
        edge_max_kernel<<<(E + 255) / 256, 256, 0, stream>>>(
            srcs[r], dsts[r], sl + (long)r * M, sr + (long)r * M, mord + (long)r * M, E);
    }
    for (int r = 0; r < 3; ++r) {
        int E = in_sizes[10 + 2 * r];
        edge_expsum_kernel<<<(E + 255) / 256, 256, 0, stream>>>(
            srcs[r], dsts[r], sl + (long)r * M, sr + (long)r * M, mord + (long)r * M,
            den + (long)r * M, E);
    }
    for (int r = 0; r < 3; ++r) {
        int E = in_sizes[10 + 2 * r];
        edge_agg_kernel<<<(E + 7) / 8, 256, 0, stream>>>(
            srcs[r], dsts[r], sl + (long)r * M, sr + (long)r * M, mord + (long)r * M,
            den + (long)r * M, Nf, hagg + (long)r * ND, E);
    }

    // q = hagg[target] @ Wq^T  (target read on device -> graph-capture safe)
    gemm_nt_wmma<<<gblocks, 256, 0, stream>>>(hagg, hagg + ND, hagg + 2 * ND, tgt,
                                              Wq, qbuf, M);
    // k_r = hagg[r] @ Wk^T
    for (int r = 0; r < 3; ++r) {
        const float* hr = hagg + (long)r * ND;
        gemm_nt_wmma<<<gblocks, 256, 0, stream>>>(hr, hr, hr, nullptr, Wk,
                                                  kbuf + (long)r * ND, M);
    }

    attn_kernel<<<(M + 7) / 8, 256, 0, stream>>>(qbuf, kbuf, hagg, (float*)d_out, M);
}